// Ops_GetPointFeat_spconv_50809463111991
// MI455X (gfx1250) — compile-verified
//
#include <hip/hip_runtime.h>

// CDNA5 / gfx1250, wave32. Distance computation is mapped onto
// V_WMMA_F32_16X16X4_F32:  D = A(16x4 voxels) x B(4x16 points) + C(|t|^2)
//   A row m = [qx, qy, qz, |q|^2]   (built on the fly from int voxel indices)
//   B col n = [-2tx, -2ty, -2tz, 1] (loop invariant per 16-point tile)
//   C[m][n] = |t_n|^2               (loop invariant accumulator input)
// => D[m][n] = |q|^2 - 2 q.t + |t|^2 = squared distance, one 16x16 tile per WMMA.
// Top-3 selection is branchless (cmp+cndmask), guarded by a wave-coherent
// "can any lane improve?" vote so stabilized tiles skip the select chain.

typedef __attribute__((ext_vector_type(2))) float v2f;
typedef __attribute__((ext_vector_type(8))) float v8f;

struct ScaleArgs {
  const int*   idx;    // [M,4] (batch, x, y, z)
  const float* feats;  // [M,C]
  int M;
  int C;
  int col_base;        // column offset in concatenated output
  float ve;            // voxel extent = 0.015 * scale
};
struct AllArgs { ScaleArgs s[4]; };

// Branchless sorted top-3 insert: (d0,i0) <= (d1,i1) <= (d2v,i2).
#define TOP3_INSERT(dv_, mi_)                                          \
  do {                                                                 \
    const float _dv = (dv_); const int _mi = (mi_);                    \
    const bool lt0 = _dv < d0;                                         \
    const bool lt1 = _dv < d1;                                         \
    const bool lt2 = _dv < d2v;                                        \
    const float nd2 = lt1 ? d1 : (lt2 ? _dv : d2v);                    \
    const int   ni2 = lt1 ? i1 : (lt2 ? _mi : i2);                     \
    const float nd1 = lt0 ? d0 : (lt1 ? _dv : d1);                     \
    const int   ni1 = lt0 ? i0 : (lt1 ? _mi : i1);                     \
    const float nd0 = lt0 ? _dv : d0;                                  \
    const int   ni0 = lt0 ? _mi : i0;                                  \
    d2v = nd2; i2 = ni2; d1 = nd1; i1 = ni1; d0 = nd0; i0 = ni0;       \
  } while (0)

__global__ __launch_bounds__(256)
void three_nn_interp_wmma(const float* __restrict__ points,  // [N,3]
                          float* __restrict__ out,           // [N, total_c]
                          AllArgs args, int n_points, int total_c)
{
  const ScaleArgs sa = args.s[blockIdx.y];
  const int lane = threadIdx.x & 31;
  const int half = lane >> 4;      // 0: lanes 0-15 (K0,K1), 1: lanes 16-31 (K2,K3)
  const int col  = lane & 15;      // point column / voxel row within 16-wide tile
  const int wavesPerBlock = blockDim.x >> 5;
  const int tile = blockIdx.x * wavesPerBlock + (threadIdx.x >> 5);
  const int p0 = tile * 16;
  if (p0 >= n_points) return;      // wave-uniform

  // ---- B operand + C accumulator (points), loop invariant --------------
  const float tx = points[(p0 + col) * 3 + 0];
  const float ty = points[(p0 + col) * 3 + 1];
  const float tz = points[(p0 + col) * 3 + 2];
  const float tsq = tx * tx + ty * ty + tz * tz;
  v2f b;
  b.x = half ? (-2.0f * tz) : (-2.0f * tx);
  b.y = half ? 1.0f         : (-2.0f * ty);
  v8f cinit;
#pragma unroll
  for (int j = 0; j < 8; ++j) cinit[j] = tsq;   // |t|^2 per column

  // ---- scan all voxel tiles, tracking per-lane top-3 smallest d^2 ------
  float d0 = 3.4e38f, d1 = 3.4e38f, d2v = 3.4e38f;
  int   i0 = 0, i1 = 0, i2 = 0;
  const float ve   = sa.ve;
  const float coff = -0.48f + 0.5f * ve;        // OFFSET + half voxel
  const int   M    = sa.M;

#pragma unroll 2
  for (int mt = 0; mt < M; mt += 16) {
    // A operand: voxel centers computed from int indices on the fly
    const int4 iv = *(const int4*)(sa.idx + (size_t)(mt + col) * 4);
    const float qx = (float)iv.y * ve + coff;
    const float qy = (float)iv.z * ve + coff;
    const float qz = (float)iv.w * ve + coff;
    const float qsq = qx * qx + qy * qy + qz * qz;
    v2f a;
    a.x = half ? qz  : qx;
    a.y = half ? qsq : qy;

    // 16x16 tile of squared distances in one matrix op (EXEC all-1s here)
    const v8f d = __builtin_amdgcn_wmma_f32_16x16x4_f32(
        false, a, false, b, (short)0, cinit, false, false);

    // Cheap per-lane min over the 8 fresh distances; skip the whole select
    // chain when no lane in the wave can improve its top-3 (single coherent
    // branch per tile; WMMA stays outside so EXEC is full when it issues).
    float m8 = fminf(fminf(fminf(d[0], d[1]), fminf(d[2], d[3])),
                     fminf(fminf(d[4], d[5]), fminf(d[6], d[7])));
    if (__any(m8 < d2v)) {
      // D layout: lane L, VGPR j -> voxel row mt + j + 8*(L>>4), col L&15
      const int mbase = mt + (half << 3);
#pragma unroll
      for (int j = 0; j < 8; ++j) {
        TOP3_INSERT(d[j], mbase + j);
      }
    }
  }

  // ---- merge half-wave top-3 sets: lane L with lane L^16 ----------------
  {
    const float e0 = __shfl_xor(d0,  16, 32);
    const float e1 = __shfl_xor(d1,  16, 32);
    const float e2 = __shfl_xor(d2v, 16, 32);
    const int   j0 = __shfl_xor(i0,  16, 32);
    const int   j1 = __shfl_xor(i1,  16, 32);
    const int   j2 = __shfl_xor(i2,  16, 32);
    TOP3_INSERT(e0, j0);
    TOP3_INSERT(e1, j1);
    TOP3_INSERT(e2, j2);
  }

  // ---- inverse-distance weights (reference: 1/(d^2 + 1e-8), normalized) -
  const float r0 = 1.0f / (d0  + 1e-8f);
  const float r1 = 1.0f / (d1  + 1e-8f);
  const float r2 = 1.0f / (d2v + 1e-8f);
  const float rs = 1.0f / (r0 + r1 + r2);
  const float w0 = r0 * rs, w1 = r1 * rs, w2 = r2 * rs;

  // ---- cooperative feature gather + blend, coalesced over channels ------
  const int C = sa.C;
  const float* __restrict__ F = sa.feats;
  float* __restrict__ obase = out + (size_t)p0 * total_c + sa.col_base;
  for (int p = 0; p < 16; ++p) {
    const int   a0 = __shfl(i0, p, 32);
    const int   a1 = __shfl(i1, p, 32);
    const int   a2 = __shfl(i2, p, 32);
    const float u0 = __shfl(w0, p, 32);
    const float u1 = __shfl(w1, p, 32);
    const float u2 = __shfl(w2, p, 32);
    const float* f0 = F + (size_t)a0 * C;
    const float* f1 = F + (size_t)a1 * C;
    const float* f2 = F + (size_t)a2 * C;
    float* op = obase + (size_t)p * total_c;
    for (int c = lane; c < C; c += 32)
      op[c] = u0 * f0[c] + u1 * f1[c] + u2 * f2[c];
  }
}

extern "C" void kernel_launch(void* const* d_in, const int* in_sizes, int n_in,
                              void* d_out, int out_size, void* d_ws, size_t ws_size,
                              hipStream_t stream) {
  // Input order (setup_inputs dict): points, batch_ids,
  //   indices1, feats1, indices2, feats2, indices3, feats3, indices4, feats4
  const float* points = (const float*)d_in[0];
  const int n_points = in_sizes[0] / 3;           // 8192
  const int SCALES[4] = {2, 4, 8, 16};

  AllArgs args;
  int col_base = 0;
  for (int i = 0; i < 4; ++i) {
    const int*   idx   = (const int*)d_in[2 + 2 * i];
    const float* feats = (const float*)d_in[3 + 2 * i];
    const int M = in_sizes[2 + 2 * i] / 4;        // [M,4]
    const int C = in_sizes[3 + 2 * i] / M;        // [M,C]
    args.s[i].idx = idx;
    args.s[i].feats = feats;
    args.s[i].M = M;
    args.s[i].C = C;
    args.s[i].col_base = col_base;
    args.s[i].ve = 0.015f * (float)SCALES[i];
    col_base += C;
  }
  const int total_c = col_base;                   // 480

  const int wavesPerBlock = 8;                    // 256 threads
  const int tiles = (n_points + 15) / 16;         // 512
  dim3 grid((tiles + wavesPerBlock - 1) / wavesPerBlock, 4);
  three_nn_interp_wmma<<<grid, dim3(256), 0, stream>>>(
      points, (float*)d_out, args, n_points, total_c);
}